// SKHamiltonian_old_5119601017176
// MI455X (gfx1250) — compile-verified
//
#include <hip/hip_runtime.h>
#include <math.h>

#ifndef __has_builtin
#define __has_builtin(x) 0
#endif

#define TILE 256
#define NF 10
#define NO 58

// Async global<->LDS movers are gfx1250-specific. Guard so compile never breaks.
#if defined(__HIP_DEVICE_COMPILE__)
#if __has_builtin(__builtin_amdgcn_global_load_async_to_lds_b128) && \
    __has_builtin(__builtin_amdgcn_global_store_async_from_lds_b128)
#define SK_HAVE_ASYNC_LDS 1
#endif
#endif

typedef int v4i __attribute__((vector_size(16)));
typedef __attribute__((address_space(1))) v4i* gv4p;   // global int4*
typedef __attribute__((address_space(3))) v4i* lv4p;   // LDS int4*

__device__ __forceinline__ gv4p as_g(const void* p) {
  return (gv4p)(unsigned long long)p;
}
__device__ __forceinline__ lv4p as_l(const void* p) {
  // Generic LDS pointer = {aperture_hi32, lds_byte_offset32}; low 32 bits are the DS address.
  return (lv4p)(unsigned int)(unsigned long long)p;
}

__device__ __forceinline__ void sk_wait_async0() {
#ifdef SK_HAVE_ASYNC_LDS
#if __has_builtin(__builtin_amdgcn_s_wait_asynccnt)
  __builtin_amdgcn_s_wait_asynccnt(0);
#else
  asm volatile("s_wait_asynccnt 0" ::: "memory");
#endif
#endif
}

// Closed-form (trig-free) Slater-Koster rotation + block assembly for one edge.
__device__ __forceinline__ void sk_compute(const float* __restrict__ f,
                                           float vx, float vy, float vz,
                                           float* __restrict__ o) {
  const float S3  = 1.7320508075688772f;   // sqrt(3)
  const float H   = 0.8660254037844386f;   // sqrt(3)/2
  const float C23 = 2.0f / 3.0f;

  float r2  = vx * vx + vy * vy + vz * vz;
  float rin = 1.0f / sqrtf(r2);
  float nx = vx * rin, ny = vy * rin, nz = vz * rin;

  float sb = sqrtf(nx * nx + ny * ny);     // sin(beta) >= 0
  float cb = nz;                           // cos(beta)
  float ca, sa;                            // cos/sin(alpha); atan2(0,0)=0 fallback
  if (sb > 1e-30f) { float ir = 1.0f / sb; ca = nx * ir; sa = ny * ir; }
  else             { ca = 1.0f; sa = 0.0f; }

  // D1[m][u] = R[(m+1)%3][(u+1)%3], R = Rz(a)*Ry(b)
  float d1[3][3] = {
    {  ca,   sa * sb, sa * cb },
    {  0.0f, cb,      -sb     },
    { -sa,   ca * sb, ca * cb }
  };

  // Columns of R
  float c0x = ca * cb, c0y = sa * cb, c0z = -sb;
  float c1x = -sa,     c1y = ca,      c1z = 0.0f;
  float c2x = ca * sb, c2y = sa * sb, c2z = cb;

  // S_u = R * A2_u * R^T; only entries {01,12,02,00,11,22} are needed.
  float S[5][6];
  // S_0 = (s3/2)(c0 c1^T + c1 c0^T)
  S[0][0] = H * (c0x * c1y + c0y * c1x);
  S[0][1] = H * (c0y * c1z + c0z * c1y);
  S[0][2] = H * (c0x * c1z + c0z * c1x);
  S[0][3] = H * 2.0f * c0x * c1x;
  S[0][4] = H * 2.0f * c0y * c1y;
  S[0][5] = H * 2.0f * c0z * c1z;
  // S_1 = (s3/2)(c1 c2^T + c2 c1^T)
  S[1][0] = H * (c1x * c2y + c1y * c2x);
  S[1][1] = H * (c1y * c2z + c1z * c2y);
  S[1][2] = H * (c1x * c2z + c1z * c2x);
  S[1][3] = H * 2.0f * c1x * c2x;
  S[1][4] = H * 2.0f * c1y * c2y;
  S[1][5] = H * 2.0f * c1z * c2z;
  // S_3 = (s3/2)(c0 c2^T + c2 c0^T)
  S[3][0] = H * (c0x * c2y + c0y * c2x);
  S[3][1] = H * (c0y * c2z + c0z * c2y);
  S[3][2] = H * (c0x * c2z + c0z * c2x);
  S[3][3] = H * 2.0f * c0x * c2x;
  S[3][4] = H * 2.0f * c0y * c2y;
  S[3][5] = H * 2.0f * c0z * c2z;
  // Outer products (single, not symmetrized)
  float O0[6] = { c0x * c0y, c0y * c0z, c0x * c0z, c0x * c0x, c0y * c0y, c0z * c0z };
  float O1[6] = { c1x * c1y, c1y * c1z, c1x * c1z, c1x * c1x, c1y * c1y, c1z * c1z };
  float O2[6] = { c2x * c2y, c2y * c2z, c2x * c2z, c2x * c2x, c2y * c2y, c2z * c2z };
#pragma unroll
  for (int i = 0; i < 6; ++i) {
    S[2][i] = O2[i] - 0.5f * (O0[i] + O1[i]);   // S_2 = -.5 c0c0 - .5 c1c1 + c2c2
    S[4][i] = H * (O0[i] - O1[i]);              // S_4 = (s3/2)(c0c0 - c1c1)
  }

  // D2[m][u] = (2/3) <A2_m, S_u>  (A2 sparsity)
  float d2[5][5];
#pragma unroll
  for (int u = 0; u < 5; ++u) {
    d2[0][u] = C23 * S3 * S[u][0];
    d2[1][u] = C23 * S3 * S[u][1];
    d2[2][u] = C23 * (S[u][5] - 0.5f * (S[u][3] + S[u][4]));
    d2[3][u] = C23 * S3 * S[u][2];
    d2[4][u] = C23 * H * (S[u][3] - S[u][4]);
  }

  // ---- blocks (f layout: ss,sp,sd, pp_s,pp_p, pd_s,pd_p, dd_s,dd_p,dd_d) ----
  o[0] = f[0];                                        // s-s
  o[1] = -f[1] * d1[0][1];                            // s-p (sign -1)
  o[2] = -f[1] * d1[1][1];
  o[3] = -f[1] * d1[2][1];
#pragma unroll
  for (int k = 0; k < 5; ++k) o[4 + k] = f[2] * d2[k][2];   // s-d (sign +1)
#pragma unroll
  for (int l = 0; l < 3; ++l)
#pragma unroll
    for (int k = 0; k < 3; ++k)                             // p-p
      o[9 + l * 3 + k] = f[3] * d1[l][1] * d1[k][1]
                       + f[4] * (d1[l][0] * d1[k][0] + d1[l][2] * d1[k][2]);
#pragma unroll
  for (int l = 0; l < 3; ++l)
#pragma unroll
    for (int k = 0; k < 5; ++k)                             // p-d (sign -1)
      o[18 + l * 5 + k] = -(f[5] * d1[l][1] * d2[k][2]
                          + f[6] * (d1[l][2] * d2[k][3] + d1[l][0] * d2[k][1]));
#pragma unroll
  for (int l = 0; l < 5; ++l)
#pragma unroll
    for (int k = 0; k < 5; ++k)                             // d-d
      o[33 + l * 5 + k] = f[7] * d2[l][2] * d2[k][2]
                        + f[8] * (d2[l][3] * d2[k][3] + d2[l][1] * d2[k][1])
                        + f[9] * (d2[l][4] * d2[k][4] + d2[l][0] * d2[k][0]);
}

__global__ __launch_bounds__(TILE) void sk_kernel(const float* __restrict__ feat,
                                                  const float* __restrict__ vec,
                                                  float* __restrict__ out, int E) {
  // One buffer, reused: input staging (13312 B) then output tile image (59392 B).
  alignas(16) __shared__ float lds[TILE * NO];
  const int tid = threadIdx.x;
  const long long base = (long long)blockIdx.x * TILE;
  const bool full = (base + TILE) <= (long long)E;

  float f[NF], vx, vy, vz;

  if (full) {
    const float* gf = feat + base * NF;   // tile-aligned: 256*40B -> 16B aligned
    const float* gv = vec + base * 3;     // 256*12B -> 16B aligned
#ifdef SK_HAVE_ASYNC_LDS
    for (int i = tid; i < (TILE * NF) / 4; i += TILE)   // 640 x b128
      __builtin_amdgcn_global_load_async_to_lds_b128(as_g(gf + i * 4),
                                                     as_l(lds + i * 4), 0, 0);
    for (int i = tid; i < (TILE * 3) / 4; i += TILE)    // 192 x b128
      __builtin_amdgcn_global_load_async_to_lds_b128(as_g(gv + i * 4),
                                                     as_l(lds + TILE * NF + i * 4), 0, 0);
    sk_wait_async0();
#else
    for (int i = tid; i < TILE * NF; i += TILE) lds[i] = gf[i];
    for (int i = tid; i < TILE * 3; i += TILE) lds[TILE * NF + i] = gv[i];
#endif
    __syncthreads();
#pragma unroll
    for (int k = 0; k < NF; ++k) f[k] = lds[tid * NF + k];  // stride 10: bank-conflict free
    vx = lds[TILE * NF + tid * 3 + 0];
    vy = lds[TILE * NF + tid * 3 + 1];
    vz = lds[TILE * NF + tid * 3 + 2];
    __syncthreads();   // LDS about to be overwritten with outputs
  } else {
    long long e = base + tid;
    if (e < (long long)E) {
#pragma unroll
      for (int k = 0; k < NF; ++k) f[k] = feat[e * NF + k];
      vx = vec[e * 3 + 0]; vy = vec[e * 3 + 1]; vz = vec[e * 3 + 2];
    } else {
#pragma unroll
      for (int k = 0; k < NF; ++k) f[k] = 0.0f;
      vx = 1.0f; vy = 0.0f; vz = 0.0f;
    }
  }

  float o[NO];
  sk_compute(f, vx, vy, vz, o);

  if (full) {
#pragma unroll
    for (int j = 0; j < NO; ++j) lds[tid * NO + j] = o[j];  // stride 58: conflict free
    __syncthreads();
    float* go = out + base * NO;   // 256*232B tile -> 16B aligned
#ifdef SK_HAVE_ASYNC_LDS
    for (int i = tid; i < (TILE * NO) / 4; i += TILE)       // 3712 x b128, lane-linear
      __builtin_amdgcn_global_store_async_from_lds_b128(as_g(go + i * 4),
                                                        as_l(lds + i * 4), 0, 0);
    sk_wait_async0();  // stores have consumed LDS before wave exit
#else
    for (int i = tid; i < TILE * NO; i += TILE) go[i] = lds[i];
#endif
  } else {
    long long e = base + tid;
    if (e < (long long)E) {
      float* go = out + e * NO;
#pragma unroll
      for (int j = 0; j < NO; ++j) go[j] = o[j];
    }
  }
}

extern "C" void kernel_launch(void* const* d_in, const int* in_sizes, int n_in,
                              void* d_out, int out_size, void* d_ws, size_t ws_size,
                              hipStream_t stream) {
  const float* feat = (const float*)d_in[0];   // (E, 10) f32
  const float* vec  = (const float*)d_in[1];   // (E, 3)  f32
  float* out = (float*)d_out;                  // (E, 58) f32
  const int E = in_sizes[0] / NF;
  const int grid = (E + TILE - 1) / TILE;
  sk_kernel<<<grid, TILE, 0, stream>>>(feat, vec, out, E);
}